// SpectralBasisEncoder_70635032150122
// MI455X (gfx1250) — compile-verified
//
#include <hip/hip_runtime.h>
#include <math.h>

// SpectralBasisEncoder fused kernel for gfx1250 (MI455X), fp32 WMMA + TDM path.
// B=16384, T=64, D=32, K=32, H=32, L=16. Memory-bound (~168MB @ 23.3TB/s => ~7us
// floor); fp32 V_WMMA_F32_16X16X4_F32 keeps reference precision with compute
// comfortably under the HBM time. Sample tiles are staged HBM->LDS with the
// Tensor Data Mover (TENSORcnt) so WMMA operands come from LDS, not long-latency
// VGPR load chains.

#define T_ 64
#define D_ 32
#define K_ 32
#define H_ 32
#define L_ 16
#define EPS_ 1e-5f
#define BT_ 16   // batch rows per workgroup

typedef __attribute__((ext_vector_type(2))) float v2f;
typedef __attribute__((ext_vector_type(8))) float v8f;
typedef __attribute__((ext_vector_type(4))) unsigned int u32x4;
typedef __attribute__((ext_vector_type(4))) int i32x4;
typedef __attribute__((ext_vector_type(8))) int i32x8;

__device__ __forceinline__ v8f wmma_f32(v2f a, v2f b, v8f c) {
  // 8 args: (neg_a, A, neg_b, B, c_mod, C, reuse_a, reuse_b)
  return __builtin_amdgcn_wmma_f32_16x16x4_f32(false, a, false, b, (short)0, c,
                                               false, false);
}

__device__ __forceinline__ unsigned rfl(unsigned x) {
  return (unsigned)__builtin_amdgcn_readfirstlane((int)x);
}

// Stage one batch row (T_*D_ = 2048 f32 = 8KB) from global into this wave's LDS
// scratch region. TDM descriptor per CDNA5 ISA:
// 2D tensor, data_size=4B, tile = 2048x1 elements.
__device__ __forceinline__ void stage_row_to_lds(const float* gsrc, float* ldst,
                                                 int lane) {
#if __has_builtin(__builtin_amdgcn_tensor_load_to_lds)
  const unsigned lds_off = rfl((unsigned)(uintptr_t)ldst);  // LDS byte offset
  const unsigned long long ga = (unsigned long long)(uintptr_t)gsrc;
  const unsigned ga_lo = rfl((unsigned)ga);
  const unsigned ga_hi = rfl((unsigned)(ga >> 32));

  // Group0: [1:0]count=1 | lds_addr[63:32] | global_addr[120:64] | type=2@[127:126]
  u32x4 g0 = {1u, lds_off, ga_lo, (ga_hi & 0x01FFFFFFu) | 0x80000000u};
  // Group1: data_size=4B@[17:16]; tensor_dim0=2048@[79:48]; tensor_dim1=1@[111:80];
  //         tile_dim0=2048@[127:112]; tile_dim1=1@[143:128];
  //         dim0_stride=2048@[207:160]; dim1_stride=2048@[255:208]
  i32x8 g1 = {(int)0x00020000u,        // workgroup_mask=0, data_size=2 (4B)
              (int)(2048u << 16),      // tensor_dim0[15:0] @ bits 63:48
              (int)(1u << 16),         // tensor_dim1[15:0] @ bits 95:80
              (int)(2048u << 16),      // tile_dim0 @ bits 127:112
              1,                       // tile_dim1=1, tile_dim2=0
              2048,                    // tensor_dim0_stride[31:0]
              (int)(2048u << 16),      // tensor_dim1_stride[15:0] @ bits 223:208
              0};
  i32x4 g2 = {0, 0, 0, 0};
  i32x4 g3 = {0, 0, 0, 0};
  i32x8 g4 = {0, 0, 0, 0, 0, 0, 0, 0};
  // clang-23 / amdgpu-toolchain form: 6 args (4 SGPR groups + pad group + cpol)
  __builtin_amdgcn_tensor_load_to_lds(g0, g1, g2, g3, g4, 0);
  __builtin_amdgcn_s_wait_tensorcnt(0);
#else
  // Fallback: coalesced b128 global loads + ds_store_b128.
  for (int c = lane; c < (T_ * D_) / 4; c += 32) {
    ((float4*)ldst)[c] = ((const float4*)gsrc)[c];
  }
  asm volatile("s_wait_dscnt 0" ::: "memory");
#endif
}

__global__ __launch_bounds__(256) void spectral_basis_encoder_kernel(
    const float* __restrict__ sample, const float* __restrict__ basis,
    const float* __restrict__ W1, const float* __restrict__ b1,
    const float* __restrict__ W2, const float* __restrict__ b2,
    const float* __restrict__ gamma, const float* __restrict__ beta,
    float* __restrict__ out) {
  __shared__ float basis_s[K_ * T_];         //  8 KB: [k][t]
  __shared__ float coeffs_s[BT_ * K_ * D_];  // 64 KB: [lb][k][d]
  __shared__ float scratch_s[8 * T_ * D_];   // 64 KB: per-wave 8KB
  // scratch_s[w]: phase A = staged sample row [t][d]; phase B = h relayout.

  const int tid  = threadIdx.x;
  const int lane = tid & 31;
  const int w    = tid >> 5;    // wave id 0..7
  const int n    = lane & 15;   // column / M-row selector within half-wave
  const int hi   = lane >> 4;   // half-wave: selects K-pair / M+8
  const int b0   = blockIdx.x * BT_;

  // Stage basis into LDS cooperatively.
  for (int i = tid; i < K_ * T_; i += 256) basis_s[i] = basis[i];
  __syncthreads();

  // ---------------- Phase A: coeffs[b] = basis[32x64] @ sample[b][64x32] ----
  // 2 passes x 8 waves: each wave TDM-stages one batch row, then runs the
  // 32x32x64 GEMM as 2x2 tiles of v_wmma_f32_16x16x4_f32 off LDS operands.
  float* ss = &scratch_s[w * (T_ * D_)];
  for (int p = 0; p < 2; ++p) {
    const int lb = p * 8 + w;  // wave-uniform
    const int gb = b0 + lb;
    stage_row_to_lds(sample + (size_t)gb * (T_ * D_), ss, lane);

    for (int dtile = 0; dtile < 2; ++dtile) {
      // B operand: 4(t) x 16(d) per step; lane n = d col, hi selects t pair.
      v2f bop[16];
#pragma unroll
      for (int s = 0; s < 16; ++s) {
        const int t0 = 4 * s + 2 * hi;
        bop[s].x = ss[(t0    ) * D_ + dtile * 16 + n];
        bop[s].y = ss[(t0 + 1) * D_ + dtile * 16 + n];
      }
      for (int ktile = 0; ktile < 2; ++ktile) {
        v8f acc = {};
#pragma unroll
        for (int s = 0; s < 16; ++s) {
          const int t0 = 4 * s + 2 * hi;
          // A operand: 16(k) x 4(t); lanes 0-15 K=t0,t0+1 / 16-31 K=t0+2,t0+3
          v2f a = *(const v2f*)&basis_s[(ktile * 16 + n) * T_ + t0];
          acc = wmma_f32(a, bop[s], acc);
        }
#pragma unroll
        for (int r = 0; r < 8; ++r) {
          const int m = r + 8 * hi;  // C layout: VGPR r -> M=r (lo) / r+8 (hi)
          coeffs_s[lb * (K_ * D_) + (ktile * 16 + m) * D_ + dtile * 16 + n] =
              acc[r];
        }
      }
    }
    // Drain this wave's LDS reads before the next pass's TDM overwrite.
    asm volatile("s_wait_dscnt 0" ::: "memory");
  }
  __syncthreads();

  // ---------------- Phase B: per-basis MLP + LayerNorm ----------------------
  // Wave w handles k in [4w, 4w+4). M dimension = the 16 local batch rows.
  float* hw = ss;  // reuse scratch region: [m][h], 2KB live
  const float g_ln = gamma[n];
  const float b_ln = beta[n];
  for (int kk = 0; kk < 4; ++kk) {
    const int k = 4 * w + kk;

    // ---- stage 2: h = gelu(coeffs[16x32] @ W1[k][32x32] + b1[k])
    v8f acc2[2];
    for (int htile = 0; htile < 2; ++htile) {
      const float bias1 = b1[k * H_ + htile * 16 + n];
#pragma unroll
      for (int r = 0; r < 8; ++r) acc2[htile][r] = bias1;
#pragma unroll
      for (int s = 0; s < 8; ++s) {
        const int d0 = 4 * s + 2 * hi;
        v2f wb;
        wb.x = W1[((size_t)k * D_ + d0    ) * H_ + htile * 16 + n];
        wb.y = W1[((size_t)k * D_ + d0 + 1) * H_ + htile * 16 + n];
        v2f a = *(const v2f*)&coeffs_s[n * (K_ * D_) + k * D_ + d0];
        acc2[htile] = wmma_f32(a, wb, acc2[htile]);
      }
    }
    // exact GELU (matches approximate=False) + relayout C->A through LDS
#pragma unroll
    for (int htile = 0; htile < 2; ++htile) {
#pragma unroll
      for (int r = 0; r < 8; ++r) {
        const float x = acc2[htile][r];
        const float gx = 0.5f * x * (1.0f + erff(x * 0.70710678118654752f));
        hw[(r + 8 * hi) * H_ + htile * 16 + n] = gx;
      }
    }
    // intra-wave LDS store->load hazard: drain DS counter
    asm volatile("s_wait_dscnt 0" ::: "memory");

    // ---- stage 3: z = h[16x32] @ W2[k][32x16] + b2[k]
    v8f acc3;
    const float bias2 = b2[k * L_ + n];
#pragma unroll
    for (int r = 0; r < 8; ++r) acc3[r] = bias2;
#pragma unroll
    for (int s = 0; s < 8; ++s) {
      const int h0 = 4 * s + 2 * hi;
      v2f wb;
      wb.x = W2[((size_t)k * H_ + h0    ) * L_ + n];
      wb.y = W2[((size_t)k * H_ + h0 + 1) * L_ + n];
      v2f a = *(const v2f*)&hw[n * H_ + h0];
      acc3 = wmma_f32(a, wb, acc3);
    }

    // ---- LayerNorm over L=16 (= N dim): butterfly within 16-lane halves
#pragma unroll
    for (int r = 0; r < 8; ++r) {
      const float z = acc3[r];
      float sum = z, sq = z * z;
#pragma unroll
      for (int m = 1; m < 16; m <<= 1) {  // masks 1,2,4,8 stay within halves
        sum += __shfl_xor(sum, m, 32);
        sq  += __shfl_xor(sq, m, 32);
      }
      const float mu  = sum * (1.0f / 16.0f);
      const float var = sq * (1.0f / 16.0f) - mu * mu;
      const float y   = g_ln * (z - mu) * rsqrtf(var + EPS_) + b_ln;
      const int gb2 = b0 + r + 8 * hi;
      out[((size_t)gb2 * K_ + k) * L_ + n] = y;
    }
  }
}

extern "C" void kernel_launch(void* const* d_in, const int* in_sizes, int n_in,
                              void* d_out, int out_size, void* d_ws,
                              size_t ws_size, hipStream_t stream) {
  const float* sample = (const float*)d_in[0];
  const float* basis  = (const float*)d_in[1];
  const float* W1     = (const float*)d_in[2];
  const float* b1     = (const float*)d_in[3];
  const float* W2     = (const float*)d_in[4];
  const float* b2     = (const float*)d_in[5];
  const float* gamma  = (const float*)d_in[6];
  const float* beta   = (const float*)d_in[7];
  float* out = (float*)d_out;

  const int Bn   = in_sizes[0] / (T_ * D_);  // 16384
  const int grid = Bn / BT_;                 // 1024 workgroups
  spectral_basis_encoder_kernel<<<grid, 256, 0, stream>>>(
      sample, basis, W1, b1, W2, b2, gamma, beta, out);
}